// TaskAlignedAssigner_CR_34608846471294
// MI455X (gfx1250) — compile-verified
//
#include <hip/hip_runtime.h>
#include <cstdint>
#include <cstddef>

// ---------------------------------------------------------------------------
// TaskAlignedAssigner for gfx1250 (MI455X, CDNA5, wave32).
// Problem sizes fixed by the reference harness.
// ---------------------------------------------------------------------------
#define BATCH 16
#define NGT   100
#define NCLS  80
#define NANCH 8400          // 80^2 + 40^2 + 20^2
#define KTOP  13
#define EPSF  1e-9f

typedef __attribute__((ext_vector_type(2))) float v2f;
typedef __attribute__((ext_vector_type(8))) float v8f;

// ---------------------------------------------------------------------------
// Kernel A: fused WMMA class-gather + IoU + alignment metric.
//   cls[b,n,l] = pred_scores[b,l,label[b,n]]  ==  onehot[n x C] @ scores^T[C x L]
// computed with V_WMMA_F32_16X16X4_F32 (exact: A is one-hot in f32).
// One wave per 16x16 (gt x anchor) tile; K=NCLS=80 -> 20 steps of K=4.
//   A layout (32-bit 16x4): lanes 0-15 = M, VGPR0 -> K=2g, VGPR1 -> K=2g+1,
//   where g = lane>>4.  D layout: VGPR j -> row M = 8g + j, col N = lane&15.
// ---------------------------------------------------------------------------
__global__ __launch_bounds__(32) void tal_align_iou_wmma(
    const float* __restrict__ pred_scores,   // [B,L,C]
    const float* __restrict__ pred_bboxes,   // [B,L,4]
    const int*   __restrict__ gt_labels,     // [B,n,1]
    const float* __restrict__ gt_bboxes,     // [B,n,4]
    const float* __restrict__ pad_mask,      // [B,n,1]
    float*       __restrict__ alignw,        // [B,n,L]
    float*       __restrict__ iouw)          // [B,n,L]
{
    const int lane = threadIdx.x;            // 0..31 (wave32)
    const int lt   = blockIdx.x;             // anchor tile (L/16 = 525)
    const int nt   = blockIdx.y;             // gt tile (ceil(100/16) = 7)
    const int b    = blockIdx.z;
    const int grp  = lane >> 4;              // 0/1 half-wave
    const int low  = lane & 15;
    const int l    = lt * 16 + low;          // this lane's anchor column

    // A fragment source: one-hot of gt label for row m = low in this gt tile
    const int grow = nt * 16 + low;
    const int lab  = (grow < NGT) ? gt_labels[b * NGT + grow] : -1;

    // B fragment source: class scores of this lane's anchor
    const float* psrow = pred_scores + ((size_t)b * NANCH + l) * NCLS;
    __builtin_prefetch(psrow, 0, 0);         // global_prefetch_b8

    v8f acc = {0.f, 0.f, 0.f, 0.f, 0.f, 0.f, 0.f, 0.f};
    #pragma unroll 4
    for (int kk = 0; kk < NCLS / 4; ++kk) {
        const int kbase = kk * 4 + 2 * grp;  // this lane covers K = kbase, kbase+1
        v2f a, bb;
        a.x  = (lab == kbase)     ? 1.0f : 0.0f;
        a.y  = (lab == kbase + 1) ? 1.0f : 0.0f;
        bb.x = psrow[kbase];
        bb.y = psrow[kbase + 1];
        acc = __builtin_amdgcn_wmma_f32_16x16x4_f32(
            /*neg_a=*/false, a, /*neg_b=*/false, bb,
            /*c_mod=*/(short)0, acc, /*reuse_a=*/false, /*reuse_b=*/false);
    }

    // Predicted box of this lane's anchor (shared across the 8 output rows)
    const float* pb = pred_bboxes + ((size_t)b * NANCH + l) * 4;
    const float px1 = pb[0], py1 = pb[1], px2 = pb[2], py2 = pb[3];
    const float parea = fmaxf(px2 - px1, 0.f) * fmaxf(py2 - py1, 0.f);

    #pragma unroll
    for (int j = 0; j < 8; ++j) {
        const int g = nt * 16 + grp * 8 + j; // D-matrix row -> gt index
        if (g >= NGT) continue;
        const float* gb = gt_bboxes + ((size_t)b * NGT + g) * 4;
        const float gx1 = gb[0], gy1 = gb[1], gx2 = gb[2], gy2 = gb[3];
        const float ix  = fmaxf(fminf(gx2, px2) - fmaxf(gx1, px1), 0.f);
        const float iy  = fmaxf(fminf(gy2, py2) - fmaxf(gy1, py1), 0.f);
        const float ov  = ix * iy;
        const float ga  = fmaxf(gx2 - gx1, 0.f) * fmaxf(gy2 - gy1, 0.f);
        const float iou = ov / (ga + parea - ov + EPSF);
        const float i2  = iou * iou;
        const float i6  = i2 * i2 * i2;       // BETA = 6, ALPHA = 1
        const float al  = acc[j] * i6 * pad_mask[b * NGT + g];
        const size_t o  = ((size_t)b * NGT + g) * NANCH + l;
        alignw[o] = al;
        iouw[o]   = iou;
    }
}

// ---------------------------------------------------------------------------
// Kernel B: per (b,gt) row: top-13 of align over L=8400 (13 argmax passes on
// an LDS copy; ties -> lowest index, matching lax.top_k). Selected anchors
// whose center lies strictly inside the gt box (d > EPS) become positives.
// ---------------------------------------------------------------------------
__global__ __launch_bounds__(256) void tal_topk_mark(
    const float* __restrict__ alignw,
    const float* __restrict__ anchor_points, // [L,2]
    const float* __restrict__ gt_bboxes,
    const float* __restrict__ pad_mask,
    unsigned char* __restrict__ maskp)       // [B,n,L], pre-zeroed
{
    const int g = blockIdx.x;
    const int b = blockIdx.y;
    if (pad_mask[b * NGT + g] == 0.f) return;            // is_in_topk *= pad

    __shared__ float sval[NANCH];                        // 33.6 KB of 320 KB LDS
    __shared__ float rv[256];
    __shared__ int   ri[256];

    const int tid = threadIdx.x;
    const size_t base = ((size_t)b * NGT + g) * NANCH;
    for (int l = tid; l < NANCH; l += 256) sval[l] = alignw[base + l];
    __syncthreads();

    const float* gb = gt_bboxes + ((size_t)b * NGT + g) * 4;
    const float gx1 = gb[0], gy1 = gb[1], gx2 = gb[2], gy2 = gb[3];

    for (int it = 0; it < KTOP; ++it) {
        float bv = -1e30f;
        int   bi = NANCH;
        for (int l = tid; l < NANCH; l += 256) {
            const float v = sval[l];
            if (v > bv || (v == bv && l < bi)) { bv = v; bi = l; }
        }
        rv[tid] = bv; ri[tid] = bi;
        __syncthreads();
        for (int off = 128; off > 0; off >>= 1) {
            if (tid < off) {
                const float ov = rv[tid + off];
                const int   oi = ri[tid + off];
                if (ov > rv[tid] || (ov == rv[tid] && oi < ri[tid])) {
                    rv[tid] = ov; ri[tid] = oi;
                }
            }
            __syncthreads();
        }
        if (tid == 0) {
            const int idx = ri[0];
            sval[idx] = -1e30f;                           // remove for next pass
            const float ax = anchor_points[2 * idx];
            const float ay = anchor_points[2 * idx + 1];
            const float d  = fminf(fminf(ax - gx1, ay - gy1),
                                   fminf(gx2 - ax, gy2 - ay));
            if (d > EPSF) maskp[base + idx] = 1;          // topk * is_in_gts
        }
        __syncthreads();
    }
}

// ---------------------------------------------------------------------------
// Kernel C: per anchor column: resolve multi-assignment (keep max-IoU gt),
// then emit labels / boxes and the assigned gt index (-1 if background).
// ---------------------------------------------------------------------------
__global__ __launch_bounds__(256) void tal_resolve(
    const float* __restrict__ iouw,
    unsigned char* __restrict__ maskp,
    const int*   __restrict__ gt_labels,
    const float* __restrict__ gt_bboxes,
    const int*   __restrict__ bg_ptr,
    float*       __restrict__ out,           // labels @0, boxes @ B*L
    int*         __restrict__ agiw)          // [B,L]
{
    const int t = blockIdx.x * blockDim.x + threadIdx.x;
    if (t >= BATCH * NANCH) return;
    const int b = t / NANCH;
    const int l = t - b * NANCH;
    const size_t base = (size_t)b * NGT * NANCH + l;

    int cnt = 0, first = -1;
    for (int g = 0; g < NGT; ++g) {
        if (maskp[base + (size_t)g * NANCH]) { if (first < 0) first = g; ++cnt; }
    }
    if (cnt > 1) {                                       // argmax of ious*mask
        float bv = -1e30f; int bmax = 0;
        for (int g = 0; g < NGT; ++g) {
            const float v = maskp[base + (size_t)g * NANCH]
                          ? iouw[base + (size_t)g * NANCH] : 0.f;
            if (v > bv) { bv = v; bmax = g; }
        }
        for (int g = 0; g < NGT; ++g)
            maskp[base + (size_t)g * NANCH] = (g == bmax) ? 1 : 0;
        first = bmax; cnt = 1;
    }

    const int agi = (first >= 0) ? first : 0;            // argmax of 0-column = 0
    const int bg  = *bg_ptr;
    const int lab = (cnt > 0) ? gt_labels[b * NGT + agi] : bg;
    out[t] = (float)lab;                                 // labels region
    const float* gb = gt_bboxes + ((size_t)b * NGT + agi) * 4;
    float* ob = out + (size_t)BATCH * NANCH + (size_t)t * 4;
    ob[0] = gb[0]; ob[1] = gb[1]; ob[2] = gb[2]; ob[3] = gb[3];
    agiw[t] = (cnt > 0) ? first : -1;
}

// ---------------------------------------------------------------------------
// Kernel D: per (b,gt) row maxima of align*mask and iou*mask.
// ---------------------------------------------------------------------------
__global__ __launch_bounds__(256) void tal_rowmax(
    const float* __restrict__ alignw,
    const float* __restrict__ iouw,
    const unsigned char* __restrict__ maskp,
    float* __restrict__ maxm,
    float* __restrict__ maxi)
{
    const int g = blockIdx.x;
    const int b = blockIdx.y;
    const size_t base = ((size_t)b * NGT + g) * NANCH;
    __shared__ float sm[256], si[256];
    const int tid = threadIdx.x;
    float mm = 0.f, mi = 0.f;                            // products are >= 0
    for (int l = tid; l < NANCH; l += 256) {
        if (maskp[base + l]) {
            mm = fmaxf(mm, alignw[base + l]);
            mi = fmaxf(mi, iouw[base + l]);
        }
    }
    sm[tid] = mm; si[tid] = mi;
    __syncthreads();
    for (int off = 128; off > 0; off >>= 1) {
        if (tid < off) {
            sm[tid] = fmaxf(sm[tid], sm[tid + off]);
            si[tid] = fmaxf(si[tid], si[tid + off]);
        }
        __syncthreads();
    }
    if (tid == 0) { maxm[b * NGT + g] = sm[0]; maxi[b * NGT + g] = si[0]; }
}

// ---------------------------------------------------------------------------
// Kernel E: per anchor: normalized score scalar scattered into its class slot.
// After resolution each column has <= 1 positive, so the n-max is direct.
// ---------------------------------------------------------------------------
__global__ __launch_bounds__(256) void tal_scores(
    const float* __restrict__ alignw,
    const int*   __restrict__ agiw,
    const float* __restrict__ maxm,
    const float* __restrict__ maxi,
    const int*   __restrict__ gt_labels,
    float*       __restrict__ out)
{
    const int t = blockIdx.x * blockDim.x + threadIdx.x;
    if (t >= BATCH * NANCH) return;
    const int b = t / NANCH;
    const int l = t - b * NANCH;
    const int g = agiw[t];
    float s = 0.f;
    int lab = -1;
    if (g >= 0) {
        const size_t o = ((size_t)b * NGT + g) * NANCH + l;
        s   = alignw[o] / (maxm[b * NGT + g] + EPSF) * maxi[b * NGT + g];
        lab = gt_labels[b * NGT + g];
    }
    float* os = out + (size_t)BATCH * NANCH * 5 + (size_t)t * NCLS;
    #pragma unroll 8
    for (int c = 0; c < NCLS; ++c) os[c] = (c == lab) ? s : 0.f;
}

// ---------------------------------------------------------------------------
extern "C" void kernel_launch(void* const* d_in, const int* in_sizes, int n_in,
                              void* d_out, int out_size, void* d_ws, size_t ws_size,
                              hipStream_t stream) {
    (void)in_sizes; (void)n_in; (void)out_size; (void)ws_size;

    const float* pred_scores   = (const float*)d_in[0];  // [B,L,C]
    const float* pred_bboxes   = (const float*)d_in[1];  // [B,L,4]
    const float* anchor_points = (const float*)d_in[2];  // [L,2]
    /* d_in[3] = stride_tensor (unused by the assigner) */
    const int*   gt_labels     = (const int*)d_in[4];    // [B,n,1]
    const float* gt_bboxes     = (const float*)d_in[5];  // [B,n,4]
    const float* pad_mask      = (const float*)d_in[6];  // [B,n,1]
    const int*   bg_ptr        = (const int*)d_in[7];    // scalar

    // Workspace layout (align + iou + mask ~= 121 MB -> fits the 192 MB L2)
    const size_t nBnL = (size_t)BATCH * NGT * NANCH;     // 13,440,000
    float*         alignw = (float*)d_ws;
    float*         iouw   = alignw + nBnL;
    unsigned char* maskp  = (unsigned char*)(iouw + nBnL);
    float*         maxm   = (float*)(maskp + nBnL);
    float*         maxi   = maxm + BATCH * NGT;
    int*           agiw   = (int*)(maxi + BATCH * NGT);
    float*         out    = (float*)d_out;

    hipMemsetAsync(maskp, 0, nBnL, stream);

    dim3 gA(NANCH / 16, (NGT + 15) / 16, BATCH);         // 525 x 7 x 16 waves
    tal_align_iou_wmma<<<gA, 32, 0, stream>>>(
        pred_scores, pred_bboxes, gt_labels, gt_bboxes, pad_mask, alignw, iouw);

    dim3 gB(NGT, BATCH);
    tal_topk_mark<<<gB, 256, 0, stream>>>(
        alignw, anchor_points, gt_bboxes, pad_mask, maskp);

    const int nBL = BATCH * NANCH;
    tal_resolve<<<(nBL + 255) / 256, 256, 0, stream>>>(
        iouw, maskp, gt_labels, gt_bboxes, bg_ptr, out, agiw);

    tal_rowmax<<<gB, 256, 0, stream>>>(alignw, iouw, maskp, maxm, maxi);

    tal_scores<<<(nBL + 255) / 256, 256, 0, stream>>>(
        alignw, agiw, maxm, maxi, gt_labels, out);
}